// BottleneckAttentionBlock_31044023616175
// MI455X (gfx1250) — compile-verified
//
#include <hip/hip_runtime.h>
#include <hip/hip_bf16.h>
#include <math.h>

#define B_    2
#define P_    2048
#define O_    512
#define D_    256
#define KSEL  3
#define NREG  2
#define H_    8
#define DH    32
#define NLAYERS 2
#define L_    2559
#define LP    2560
#define N_QKV 768

typedef float v2f __attribute__((ext_vector_type(2)));
typedef float v8f __attribute__((ext_vector_type(8)));
typedef int   v4i __attribute__((ext_vector_type(4)));

static __device__ __forceinline__ v8f wmma_f32(v2f a, v2f b, v8f c) {
  // V_WMMA_F32_16X16X4_F32: (neg_a, A, neg_b, B, c_mod, C, reuse_a, reuse_b)
  return __builtin_amdgcn_wmma_f32_16x16x4_f32(false, a, false, b, (short)0, c,
                                               false, false);
}

// --------- gfx1250 async global->LDS staging (ASYNCcnt path), with fallback
#if defined(__has_builtin)
#if __has_builtin(__builtin_amdgcn_global_load_async_to_lds_b128) && \
    __has_builtin(__builtin_amdgcn_s_wait_asynccnt)
#define HAVE_ASYNC_LDS 1
#endif
#endif

#ifdef HAVE_ASYNC_LDS
// Signature (from hipcc diagnostics): (v4i AS1* gsrc, v4i AS3* ldst, Imm, Imm)
#define ASYNC_CP16(gsrc, ldst)                                                 \
  __builtin_amdgcn_global_load_async_to_lds_b128(                              \
      (__attribute__((address_space(1))) v4i*)(gsrc),                          \
      (__attribute__((address_space(3))) v4i*)(ldst), 0, 0)
#define ASYNC_WAIT() __builtin_amdgcn_s_wait_asynccnt(0)
#else
#define ASYNC_CP16(gsrc, ldst)                                                 \
  do { *(float4*)(ldst) = *(const float4*)(gsrc); } while (0)
#define ASYNC_WAIT() do {} while (0)
#endif

// ---------------------------------------------------------------- row norms
__global__ __launch_bounds__(256) void norms_kernel(const float* __restrict__ X,
                                                    float* __restrict__ out,
                                                    int rows) {
  int row  = blockIdx.x * 8 + (threadIdx.x >> 5);
  int lane = threadIdx.x & 31;
  if (row >= rows) return;
  const float* xr = X + (size_t)row * D_;
  float4 a = ((const float4*)xr)[lane];
  float4 c = ((const float4*)xr)[lane + 32];
  float s = a.x*a.x + a.y*a.y + a.z*a.z + a.w*a.w +
            c.x*c.x + c.y*c.y + c.z*c.z + c.w*c.w;
#pragma unroll
  for (int off = 16; off > 0; off >>= 1) s += __shfl_xor(s, off, 32);
  if (lane == 0) out[row] = fmaxf(sqrtf(s), 1e-8f);
}

// ------------------------------------------------- cosine-sim GEMM (WMMA f32)
__global__ __launch_bounds__(128) void sim_kernel(const float* __restrict__ xp,
                                                  const float* __restrict__ xo,
                                                  const float* __restrict__ pn,
                                                  const float* __restrict__ on_,
                                                  float* __restrict__ sim) {
  __shared__ float As[32][260];                 // row base 1040B, 16B aligned
  int b  = blockIdx.z;
  int ot = blockIdx.x * 32, pt = blockIdx.y * 32;
  const float* xpb = xp + (size_t)b * P_ * D_;
  const float* xob = xo + (size_t)b * O_ * D_;
  for (int i = threadIdx.x; i < 2048; i += 128) {
    int r = i >> 6, c4 = i & 63;
    ASYNC_CP16(xpb + (size_t)(pt + r) * D_ + c4 * 4, &As[r][c4 * 4]);
  }
  ASYNC_WAIT();
  __syncthreads();
  int wave = threadIdx.x >> 5, lane = threadIdx.x & 31;
  int wr = wave >> 1, wc = wave & 1;
  int m0 = wr * 16, n0 = wc * 16;
  int nn = lane & 15, hf = lane >> 4;
  int go = ot + n0 + nn;
  const float* xoc = xob + (size_t)go * D_;
  v8f acc = {};
  for (int k0 = 0; k0 < D_; k0 += 4) {
    int ka = k0 + 2 * hf;
    v2f a;  a[0]  = As[m0 + nn][ka]; a[1]  = As[m0 + nn][ka + 1];
    v2f bb; bb[0] = xoc[ka];         bb[1] = xoc[ka + 1];
    acc = wmma_f32(a, bb, acc);
  }
  float no = on_[b * O_ + go];
#pragma unroll
  for (int r = 0; r < 8; ++r) {
    int gp = pt + m0 + r + 8 * hf;
    sim[((size_t)b * P_ + gp) * O_ + go] = acc[r] / (pn[b * P_ + gp] * no);
  }
}

// ------------------------------------------------------------- greedy argmax
__global__ __launch_bounds__(512) void greedy_kernel(const float* __restrict__ sim,
                                                     int* __restrict__ sel) {
  int b = blockIdx.x;
  const float* s = sim + (size_t)b * P_ * O_;
  __shared__ int   sip[3], sio[3];
  __shared__ float wv[16];
  __shared__ int   wi[16];
  int t = threadIdx.x, lane = t & 31, wave = t >> 5;
  if (t < 3) { sip[t] = -1; sio[t] = -1; }
  __syncthreads();
  for (int kk = 0; kk < KSEL; ++kk) {
    int p0 = sip[0], p1 = sip[1], p2 = sip[2];
    int q0 = sio[0], q1 = sio[1], q2 = sio[2];
    float bv = -INFINITY; int bi = 0x7fffffff;
    for (int idx = t; idx < P_ * O_; idx += 512) {
      int p = idx >> 9, o = idx & (O_ - 1);
      if (p == p0 || p == p1 || p == p2 || o == q0 || o == q1 || o == q2) continue;
      float v = s[idx];
      if (v > bv || (v == bv && idx < bi)) { bv = v; bi = idx; }
    }
#pragma unroll
    for (int off = 16; off > 0; off >>= 1) {
      float ov = __shfl_xor(bv, off, 32);
      int   oi = __shfl_xor(bi, off, 32);
      if (ov > bv || (ov == bv && oi < bi)) { bv = ov; bi = oi; }
    }
    if (lane == 0) { wv[wave] = bv; wi[wave] = bi; }
    __syncthreads();
    if (t < 16) {
      bv = wv[t]; bi = wi[t];
#pragma unroll
      for (int off = 8; off > 0; off >>= 1) {
        float ov = __shfl_xor(bv, off, 32);
        int   oi = __shfl_xor(bi, off, 32);
        if (ov > bv || (ov == bv && oi < bi)) { bv = ov; bi = oi; }
      }
      if (t == 0) {
        int ip = bi >> 9, io = bi & (O_ - 1);
        sip[kk] = ip; sio[kk] = io;
        sel[b * 8 + kk]     = ip;
        sel[b * 8 + 4 + kk] = io;
      }
    }
    __syncthreads();
  }
}

// ------------------------------------------------------------- paired tokens
__global__ __launch_bounds__(256) void ks_kernel(const float* __restrict__ xp,
                                                 const float* __restrict__ xo,
                                                 const float* __restrict__ Wlp,
                                                 const float* __restrict__ blp,
                                                 const float* __restrict__ Wlo,
                                                 const float* __restrict__ blo,
                                                 const int* __restrict__ sel,
                                                 float* __restrict__ x) {
  int kk = blockIdx.x, b = blockIdx.y, t = threadIdx.x;
  int ip = sel[b * 8 + kk], io = sel[b * 8 + 4 + kk];
  const float* xpr  = xp + ((size_t)b * P_ + ip) * D_;
  const float* xor_ = xo + ((size_t)b * O_ + io) * D_;
  float acc = blp[t] + blo[t];
  for (int d = 0; d < D_; ++d)
    acc += xpr[d] * Wlp[d * D_ + t] + xor_[d] * Wlo[d * D_ + t];
  x[((size_t)b * LP + kk) * D_ + t] = acc;
}

// ------------------------------------------------------------------ assembly
__global__ __launch_bounds__(64) void assemble_kernel(const float* __restrict__ xp,
                                                      const float* __restrict__ xo,
                                                      const float* __restrict__ bottle,
                                                      const int* __restrict__ sel,
                                                      float* __restrict__ x) {
  int sidx = blockIdx.x, b = blockIdx.y, t = threadIdx.x;
  const int* sb = sel + b * 8;
  const float* src; int dest;
  if (sidx < P_) {
    int cnt = 0;
#pragma unroll
    for (int j = 0; j < 3; ++j) { if (sb[j] == sidx) return; if (sb[j] < sidx) cnt++; }
    dest = KSEL + sidx - cnt;
    src  = xp + ((size_t)b * P_ + sidx) * D_;
  } else if (sidx < P_ + O_) {
    int i = sidx - P_;
    int cnt = 0;
#pragma unroll
    for (int j = 0; j < 3; ++j) { if (sb[4+j] == i) return; if (sb[4+j] < i) cnt++; }
    dest = P_ + NREG + i - cnt;                 // 2050 + i - cnt
    src  = xo + ((size_t)b * O_ + i) * D_;
  } else if (sidx < P_ + O_ + NREG) {
    int j = sidx - P_ - O_;
    dest = P_ + j;                              // 2048 + j
    src  = bottle + (size_t)j * D_;
  } else {                                      // pad row: zero it
    float4 z = {0.f, 0.f, 0.f, 0.f};
    ((float4*)(x + ((size_t)b * LP + (LP - 1)) * D_))[t] = z;
    return;
  }
  float4 v = ((const float4*)src)[t];
  ((float4*)(x + ((size_t)b * LP + dest) * D_))[t] = v;
}

// ----------------------------------------------------------------- layernorm
__global__ __launch_bounds__(256) void ln_kernel(const float* __restrict__ x,
                                                 const float* __restrict__ g,
                                                 const float* __restrict__ be,
                                                 float* __restrict__ y, int rows) {
  int row  = blockIdx.x * 8 + (threadIdx.x >> 5);
  int lane = threadIdx.x & 31;
  if (row >= rows) return;
  const float* xr = x + (size_t)row * D_;
  float4 a = ((const float4*)xr)[lane];
  float4 c = ((const float4*)xr)[lane + 32];
  float s = a.x + a.y + a.z + a.w + c.x + c.y + c.z + c.w;
#pragma unroll
  for (int off = 16; off > 0; off >>= 1) s += __shfl_xor(s, off, 32);
  float mu = s * (1.0f / 256.0f);
  float d0 = a.x - mu, d1 = a.y - mu, d2 = a.z - mu, d3 = a.w - mu;
  float e0 = c.x - mu, e1 = c.y - mu, e2 = c.z - mu, e3 = c.w - mu;
  float v = d0*d0 + d1*d1 + d2*d2 + d3*d3 + e0*e0 + e1*e1 + e2*e2 + e3*e3;
#pragma unroll
  for (int off = 16; off > 0; off >>= 1) v += __shfl_xor(v, off, 32);
  float inv = rsqrtf(v * (1.0f / 256.0f) + 1e-5f);
  float4 g0 = ((const float4*)g)[lane],  g1 = ((const float4*)g)[lane + 32];
  float4 b0 = ((const float4*)be)[lane], b1 = ((const float4*)be)[lane + 32];
  float4 o0, o1;
  o0.x = d0*inv*g0.x + b0.x; o0.y = d1*inv*g0.y + b0.y;
  o0.z = d2*inv*g0.z + b0.z; o0.w = d3*inv*g0.w + b0.w;
  o1.x = e0*inv*g1.x + b1.x; o1.y = e1*inv*g1.y + b1.y;
  o1.z = e2*inv*g1.z + b1.z; o1.w = e3*inv*g1.w + b1.w;
  ((float4*)(y + (size_t)row * D_))[lane]      = o0;
  ((float4*)(y + (size_t)row * D_))[lane + 32] = o1;
}

// ---------------------------------- generic K=256 GEMM: C = A*W + bias (WMMA)
__global__ __launch_bounds__(256) void gemm_kernel(const float* __restrict__ A,
                                                   const float* __restrict__ W,
                                                   const float* __restrict__ bias,
                                                   float* __restrict__ C,
                                                   int N, int addInto, int skipPad) {
  __shared__ float As[32][260];
  int tile_n = blockIdx.x * 64;
  int tile_m = blockIdx.y * 32;
  // Warm L2/WGP$ with the 256-row W panel this block will consume.
  __builtin_prefetch(W + (size_t)threadIdx.x * N + tile_n, 0, 1);
  for (int i = threadIdx.x; i < 2048; i += 256) {
    int r = i >> 6, c4 = i & 63;
    ASYNC_CP16(A + (size_t)(tile_m + r) * D_ + c4 * 4, &As[r][c4 * 4]);
  }
  ASYNC_WAIT();
  __syncthreads();
  int wave = threadIdx.x >> 5, lane = threadIdx.x & 31;
  int wr = wave >> 2, wc = wave & 3;
  int m0 = wr * 16;
  int nn = lane & 15, hf = lane >> 4;
  int ncol = tile_n + wc * 16 + nn;
  v8f acc = {};
  for (int k0 = 0; k0 < D_; k0 += 4) {
    int ka = k0 + 2 * hf;
    v2f a; a[0] = As[m0 + nn][ka]; a[1] = As[m0 + nn][ka + 1];
    const float* wp = W + (size_t)ka * N + ncol;
    v2f bb; bb[0] = wp[0]; bb[1] = wp[N];
    acc = wmma_f32(a, bb, acc);
  }
  float bv = bias[ncol];
#pragma unroll
  for (int r = 0; r < 8; ++r) {
    int grow = tile_m + m0 + r + 8 * hf;
    if (skipPad && (grow % LP) == (LP - 1)) continue;   // keep pad row of x at 0
    size_t o = (size_t)grow * N + ncol;
    float val = acc[r] + bv;
    if (addInto) val += C[o];
    C[o] = val;
  }
}

// --------------------------------------- flash attention, one wave per q-tile
__global__ __launch_bounds__(32) void attn_kernel(const float* __restrict__ qkv,
                                                  float* __restrict__ ao) {
  __shared__ float Qs[16][36];    // row 144B: 16B aligned, stride 36 mod 64 banks
  __shared__ float KVs[16][68];   // row 272B: 16B aligned, stride 4 mod 64 banks
  __shared__ float Ps[16][17];
  __shared__ float arow[16];
  __shared__ float lrow[16];
  int qt = blockIdx.x, h = blockIdx.y, b = blockIdx.z;
  int lane = threadIdx.x;
  int nn = lane & 15, hf = lane >> 4;
  int q0 = qt * 16;
  const float* base = qkv + (size_t)b * LP * N_QKV;
  for (int i = lane; i < 16 * 8; i += 32) {           // Q tile 16x32
    int r = i >> 3, c = i & 7;
    ASYNC_CP16(base + (size_t)(q0 + r) * N_QKV + h * DH + c * 4, &Qs[r][c * 4]);
  }
  ASYNC_WAIT();
  v8f acc0 = {}, acc1 = {};
  float mcur = -INFINITY, lcur = 0.0f;
  const float scale = 0.17677669529663687f;           // 1/sqrt(32)
  for (int kt = 0; kt < LP / 16; ++kt) {
    int k0 = kt * 16;
    __syncthreads();
    for (int i = lane; i < 16 * 16; i += 32) {        // K|V tile 16x(32+32)
      int r = i >> 4, c = i & 15;
      const float* src = base + (size_t)(k0 + r) * N_QKV +
          ((c < 8) ? (D_ + h * DH + c * 4) : (2 * D_ + h * DH + (c - 8) * 4));
      ASYNC_CP16(src, &KVs[r][c * 4]);
    }
    ASYNC_WAIT();
    __syncthreads();
    v8f s = {};
#pragma unroll
    for (int kk = 0; kk < DH; kk += 4) {              // S = Q K^T
      int ka = kk + 2 * hf;
      v2f a;  a[0]  = Qs[nn][ka];  a[1]  = Qs[nn][ka + 1];
      v2f bb; bb[0] = KVs[nn][ka]; bb[1] = KVs[nn][ka + 1];
      s = wmma_f32(a, bb, s);
    }
#pragma unroll
    for (int r = 0; r < 8; ++r) Ps[r + 8 * hf][nn] = s[r];
    __syncthreads();
    if (lane < 16) {                                  // online softmax, row=lane
      float mx = -INFINITY;
#pragma unroll
      for (int j = 0; j < 16; ++j) {
        float sv = (k0 + j < L_) ? Ps[lane][j] * scale : -INFINITY;
        Ps[lane][j] = sv;
        mx = fmaxf(mx, sv);
      }
      float mnew  = fmaxf(mcur, mx);
      float alpha = expf(mcur - mnew);
      float lnew  = lcur * alpha;
#pragma unroll
      for (int j = 0; j < 16; ++j) {
        float p = (k0 + j < L_) ? expf(Ps[lane][j] - mnew) : 0.0f;
        Ps[lane][j] = p;
        lnew += p;
      }
      mcur = mnew; lcur = lnew;
      arow[lane] = alpha;
    }
    __syncthreads();
#pragma unroll
    for (int r = 0; r < 8; ++r) {
      float f = arow[r + 8 * hf];
      acc0[r] *= f; acc1[r] *= f;
    }
#pragma unroll
    for (int kk = 0; kk < 16; kk += 4) {              // O += P @ V
      int ka = kk + 2 * hf;
      v2f a;  a[0]  = Ps[nn][ka];        a[1]  = Ps[nn][ka + 1];
      v2f b0; b0[0] = KVs[ka][32 + nn];  b0[1] = KVs[ka + 1][32 + nn];
      v2f b1; b1[0] = KVs[ka][48 + nn];  b1[1] = KVs[ka + 1][48 + nn];
      acc0 = wmma_f32(a, b0, acc0);
      acc1 = wmma_f32(a, b1, acc1);
    }
  }
  __syncthreads();
  if (lane < 16) lrow[lane] = lcur;
  __syncthreads();
  float* outb = ao + ((size_t)b * LP + q0) * D_ + h * DH;
#pragma unroll
  for (int r = 0; r < 8; ++r) {
    int row = r + 8 * hf;
    float inv = 1.0f / lrow[row];
    outb[(size_t)row * D_ + nn]      = acc0[r] * inv;
    outb[(size_t)row * D_ + 16 + nn] = acc1[r] * inv;
  }
}

// ------------------------------------------------------------- output gather
__global__ __launch_bounds__(64) void output_kernel(const float* __restrict__ x,
                                                    float* __restrict__ out) {
  int idx = blockIdx.x;
  int b = idx / 2557, r = idx % 2557;
  size_t srow, drow;
  if (r == 0)        { srow = (size_t)b * LP;               drow = (size_t)b; }
  else if (r < P_)   { srow = (size_t)b * LP + r;
                       drow = (size_t)B_ + (size_t)b * (P_ - 1) + (r - 1); }
  else if (r == P_)  { srow = (size_t)b * LP + (P_ + NREG);
                       drow = (size_t)B_ + (size_t)B_ * (P_ - 1) + b; }
  else { int i = r - P_ - 1;
         srow = (size_t)b * LP + P_ + NREG + 1 + i;
         drow = (size_t)B_ + (size_t)B_ * (P_ - 1) + B_ +
                (size_t)b * (O_ - KSEL - 1) + i; }
  float4 v = ((const float4*)(x + srow * D_))[threadIdx.x];
  ((float4*)(out + drow * D_))[threadIdx.x] = v;
}

// =============================================================== host launch
extern "C" void kernel_launch(void* const* d_in, const int* in_sizes, int n_in,
                              void* d_out, int out_size, void* d_ws, size_t ws_size,
                              hipStream_t stream) {
  (void)in_sizes; (void)n_in; (void)out_size; (void)ws_size;
  const float* xp     = (const float*)d_in[0];
  const float* xo     = (const float*)d_in[1];
  const float* bottle = (const float*)d_in[2];
  const float* Wlp    = (const float*)d_in[3];
  const float* blp    = (const float*)d_in[4];
  const float* Wlo    = (const float*)d_in[5];
  const float* blo    = (const float*)d_in[6];
  const float* ln_g   = (const float*)d_in[7];
  const float* ln_b   = (const float*)d_in[8];
  const float* Wqkv   = (const float*)d_in[9];
  const float* bqkv   = (const float*)d_in[10];
  const float* Wo     = (const float*)d_in[11];
  const float* bo     = (const float*)d_in[12];
  float* out = (float*)d_out;

  float* ws = (float*)d_ws;
  size_t off = 0;
  float* pn  = ws + off; off += (size_t)B_ * P_;
  float* on_ = ws + off; off += (size_t)B_ * O_;
  int*   sel = (int*)(ws + off); off += 16;
  float* sim = ws + off; off += (size_t)B_ * P_ * O_;
  float* x   = ws + off; off += (size_t)B_ * LP * D_;
  float* y   = ws + off; off += (size_t)B_ * LP * D_;
  float* qkv = ws + off; off += (size_t)B_ * LP * N_QKV;
  float* ao  = ws + off; off += (size_t)B_ * LP * D_;

  norms_kernel<<<(B_ * P_) / 8, 256, 0, stream>>>(xp, pn, B_ * P_);
  norms_kernel<<<(B_ * O_) / 8, 256, 0, stream>>>(xo, on_, B_ * O_);
  sim_kernel<<<dim3(O_ / 32, P_ / 32, B_), 128, 0, stream>>>(xp, xo, pn, on_, sim);
  greedy_kernel<<<B_, 512, 0, stream>>>(sim, sel);
  ks_kernel<<<dim3(KSEL, B_), 256, 0, stream>>>(xp, xo, Wlp, blp, Wlo, blo, sel, x);
  assemble_kernel<<<dim3(P_ + O_ + NREG + 1, B_), 64, 0, stream>>>(xp, xo, bottle, sel, x);

  for (int l = 0; l < NLAYERS; ++l) {
    ln_kernel<<<(B_ * LP) / 8, 256, 0, stream>>>(x, ln_g + l * D_, ln_b + l * D_,
                                                 y, B_ * LP);
    gemm_kernel<<<dim3(N_QKV / 64, (B_ * LP) / 32), 256, 0, stream>>>(
        y, Wqkv + (size_t)l * D_ * N_QKV, bqkv + (size_t)l * N_QKV, qkv,
        N_QKV, 0, 0);
    attn_kernel<<<dim3(LP / 16, H_, B_), 32, 0, stream>>>(qkv, ao);
    gemm_kernel<<<dim3(D_ / 64, (B_ * LP) / 32), 256, 0, stream>>>(
        ao, Wo + (size_t)l * D_ * D_, bo + (size_t)l * D_, x, D_, 1, 1);
  }
  output_kernel<<<B_ * 2557, 64, 0, stream>>>(x, out);
}